// GemmaAttention_57406532878472
// MI455X (gfx1250) — compile-verified
//
#include <hip/hip_runtime.h>
#include <math.h>

typedef float v2f __attribute__((ext_vector_type(2)));
typedef float v8f __attribute__((ext_vector_type(8)));

#define BB   32
#define HIDW 3072
#define NH   16
#define NKV  8
#define HD   256
#define SEQ  2048
#define QKN  (NH * HD)           // 4096
#define KVN  (NKV * HD)          // 2048
#define YCOLS (QKN + 2 * KVN)    // 8192
#define SCALE 0.0625f            // 256^-0.5

// ---------------------------------------------------------------------------
// C[32 x N] = A[32 x K] @ W[K x N], fp32, via V_WMMA_F32_16X16X4_F32.
// Compile-time N/K/LDC so all row strides fold into the 24-bit immediate
// offset of global_load_* (no per-iteration 64-bit address math).
// One wave per 16x16 output tile; 8 waves per block share the M-tile so
// adjacent N-tiles share W cachelines. 4 accumulators break the WMMA RAW
// dependency chain (ISA 7.12.1). Fragment layout per ISA 7.12.2 (32-bit A
// 16x4): lane<16 -> K={0,1}, lane>=16 -> K={2,3}; B mirrored; C/D rows
// split at lane 16.
// ---------------------------------------------------------------------------
template <int N, int K, int LDC>
__global__ __launch_bounds__(256)
void gemm32_wmma_f32(const float* __restrict__ A, const float* __restrict__ W,
                     float* __restrict__ C) {
  constexpr int NTILE = N >> 4;
  const int lane = threadIdx.x & 31;
  const int wave = threadIdx.x >> 5;
  const int tile = blockIdx.x * 8 + wave;
  const int mt   = tile / NTILE;
  const int nt   = tile - mt * NTILE;
  const int r16  = lane & 15;
  const int khi  = lane >> 4;             // 0 or 1

  const float* __restrict__ Ap = A + (size_t)(mt * 16 + r16) * K + 2 * khi;
  const int    ncol            = nt * 16 + r16;
  const float* __restrict__ Bp = W + (size_t)(2 * khi) * N + ncol;

  v8f acc0 = {}, acc1 = {}, acc2 = {}, acc3 = {};

  for (int k0 = 0; k0 < K; k0 += 16) {
    // Speculative prefetch of the weight stream ~64 k-rows ahead
    // (global_prefetch_b8; silently dropped past the end of the buffer).
    __builtin_prefetch(Bp + (size_t)64 * N, 0, 1);
#pragma unroll
    for (int j = 0; j < 4; ++j) {
      v2f a, b;
      a.x = Ap[4 * j];                    // immediate offsets, b64-merged
      a.y = Ap[4 * j + 1];
      b.x = Bp[(size_t)(4 * j) * N];      // immediate offsets (const N)
      b.y = Bp[(size_t)(4 * j) * N + N];
      v8f* accp = (j == 0) ? &acc0 : (j == 1) ? &acc1 : (j == 2) ? &acc2
                                                                 : &acc3;
      *accp = __builtin_amdgcn_wmma_f32_16x16x4_f32(
          /*neg_a=*/false, a, /*neg_b=*/false, b,
          /*c_mod=*/(short)0, *accp, /*reuse_a=*/false, /*reuse_b=*/false);
    }
    Ap += 16;
    Bp += (size_t)16 * N;
  }

  v8f res = (acc0 + acc1) + (acc2 + acc3);
  const int mbase = mt * 16 + khi * 8;
#pragma unroll
  for (int r = 0; r < 8; ++r)
    C[(size_t)(mbase + r) * LDC + ncol] = res[r];
}

// ---------------------------------------------------------------------------
// RoPE on q (cols 0..4095) and k (cols 4096..6143), pass-through v.
// Reference: out[:128] = xr*c - xi*s ; out[128:] = xr*s + xi*c  (per head).
// ---------------------------------------------------------------------------
__global__ __launch_bounds__(256)
void rope_split(const float* __restrict__ Y,
                const float* __restrict__ fc, const float* __restrict__ fs,
                float* __restrict__ qws, float* __restrict__ kws,
                float* __restrict__ vws) {
  const int idx = blockIdx.x * blockDim.x + threadIdx.x;  // < 32*8192
  const int b   = idx >> 13;
  const int col = idx & (YCOLS - 1);
  const float x = Y[idx];

  if (col < QKN + KVN) {
    const int d = col & (HD - 1);
    float out;
    if (d < 128) {
      const float c  = fc[b * 128 + d];
      const float s  = fs[b * 128 + d];
      const float xi = Y[idx + 128];
      out = x * c - xi * s;
    } else {
      const int dd   = d - 128;
      const float c  = fc[b * 128 + dd];
      const float s  = fs[b * 128 + dd];
      const float xr = Y[idx - 128];
      out = xr * s + x * c;
    }
    if (col < QKN) qws[(size_t)b * QKN + col] = out;
    else           kws[(size_t)b * KVN + (col - QKN)] = out;
  } else {
    vws[(size_t)b * KVN + (col - QKN - KVN)] = x;
  }
}

// ---------------------------------------------------------------------------
// Attention: one block per (b, kv); 256 threads. G=2 query rows.
// The cache row at s==pos is replaced by the freshly computed k/v (inputs
// are immutable). Phase 1: scores (thread-per-key, q in LDS). Phase 2:
// block softmax. Phase 3: P@V with coalesced V reads + LDS partial reduce.
// ---------------------------------------------------------------------------
__global__ __launch_bounds__(256)
void attn_decode(const float* __restrict__ qws, const float* __restrict__ kws,
                 const float* __restrict__ vws,
                 const float* __restrict__ cache_k,
                 const float* __restrict__ cache_v,
                 const float* __restrict__ mask,
                 float* __restrict__ attn_out, const int* __restrict__ pos_p) {
  const int b   = blockIdx.x >> 3;
  const int kv  = blockIdx.x & 7;
  const int tid = threadIdx.x;
  const int pos = *pos_p;

  __shared__ float sQ[2][HD];        // 2 KB
  __shared__ float sScore[2][SEQ];   // 16 KB
  __shared__ float sRed[256];        // 1 KB
  __shared__ float sOut[4][2][HD];   // 8 KB

  // load both query rows (h = kv*2 + g) into LDS
#pragma unroll
  for (int g = 0; g < 2; ++g)
    sQ[g][tid] = qws[(size_t)b * QKN + (size_t)(kv * 2 + g) * HD + tid];
  __syncthreads();

  const float* __restrict__ Kbase  = cache_k + ((size_t)b * NKV + kv) * SEQ * HD;
  const float* __restrict__ Kfresh = kws + (size_t)b * KVN + (size_t)kv * HD;

  // ---- Phase 1: scores[g][s] = SCALE * q_g . k_s + mask[b][s]
#pragma unroll
  for (int i = 0; i < 8; ++i) {
    const int s = tid + (i << 8);
    const bool fresh = (s >= pos) && (s < pos + 1);
    const float4* __restrict__ kr =
        (const float4*)(fresh ? Kfresh : (Kbase + (size_t)s * HD));
    float acc0 = 0.f, acc1 = 0.f;
#pragma unroll 4
    for (int d4 = 0; d4 < HD / 4; ++d4) {
      const float4 kk = kr[d4];
      const float4 q0 = ((const float4*)sQ[0])[d4];
      const float4 q1 = ((const float4*)sQ[1])[d4];
      acc0 += kk.x * q0.x + kk.y * q0.y + kk.z * q0.z + kk.w * q0.w;
      acc1 += kk.x * q1.x + kk.y * q1.y + kk.z * q1.z + kk.w * q1.w;
    }
    const float mval = mask[(size_t)b * SEQ + s];
    sScore[0][s] = acc0 * SCALE + mval;
    sScore[1][s] = acc1 * SCALE + mval;
  }
  __syncthreads();

  // ---- Phase 2: softmax per row
#pragma unroll
  for (int g = 0; g < 2; ++g) {
    float lm = -INFINITY;
#pragma unroll
    for (int i = 0; i < 8; ++i) lm = fmaxf(lm, sScore[g][tid + (i << 8)]);
    sRed[tid] = lm;
    __syncthreads();
    for (int off = 128; off > 0; off >>= 1) {
      if (tid < off) sRed[tid] = fmaxf(sRed[tid], sRed[tid + off]);
      __syncthreads();
    }
    const float gm = sRed[0];
    __syncthreads();

    float ls = 0.f;
#pragma unroll
    for (int i = 0; i < 8; ++i) {
      const int s   = tid + (i << 8);
      const float e = __expf(sScore[g][s] - gm);
      sScore[g][s]  = e;
      ls += e;
    }
    sRed[tid] = ls;
    __syncthreads();
    for (int off = 128; off > 0; off >>= 1) {
      if (tid < off) sRed[tid] += sRed[tid + off];
      __syncthreads();
    }
    const float inv = 1.0f / sRed[0];
    __syncthreads();
#pragma unroll
    for (int i = 0; i < 8; ++i) sScore[g][tid + (i << 8)] *= inv;
    __syncthreads();
  }

  // ---- Phase 3: out[g][d] = sum_s p[g][s] * v[s][d]
  const int dgrp = tid & 63;        // d = 4*dgrp
  const int sgrp = tid >> 6;        // 0..3
  const float* __restrict__ Vbase  = cache_v + ((size_t)b * NKV + kv) * SEQ * HD;
  const float* __restrict__ Vfresh = vws + (size_t)b * KVN + (size_t)kv * HD;

  float4 o0 = {0.f, 0.f, 0.f, 0.f}, o1 = {0.f, 0.f, 0.f, 0.f};
  for (int i = 0; i < SEQ / 4; ++i) {
    const int s = (i << 2) + sgrp;
    const bool fresh = (s >= pos) && (s < pos + 1);
    const float4 vv =
        ((const float4*)(fresh ? Vfresh : (Vbase + (size_t)s * HD)))[dgrp];
    const float p0 = sScore[0][s];
    const float p1 = sScore[1][s];
    o0.x += vv.x * p0; o0.y += vv.y * p0; o0.z += vv.z * p0; o0.w += vv.w * p0;
    o1.x += vv.x * p1; o1.y += vv.y * p1; o1.z += vv.z * p1; o1.w += vv.w * p1;
  }
  ((float4*)&sOut[sgrp][0][0])[dgrp] = o0;
  ((float4*)&sOut[sgrp][1][0])[dgrp] = o1;
  __syncthreads();

  // reduce the 4 s-group partials; thread tid owns dimension d = tid
#pragma unroll
  for (int g = 0; g < 2; ++g) {
    const float v = (sOut[0][g][tid] + sOut[1][g][tid]) +
                    (sOut[2][g][tid] + sOut[3][g][tid]);
    attn_out[(size_t)b * QKN + (size_t)(kv * 2 + g) * HD + tid] = v;
  }
}

// ---------------------------------------------------------------------------
extern "C" void kernel_launch(void* const* d_in, const int* in_sizes, int n_in,
                              void* d_out, int out_size, void* d_ws,
                              size_t ws_size, hipStream_t stream) {
  const float* hidden = (const float*)d_in[0];
  const float* fcos   = (const float*)d_in[1];
  const float* fsin   = (const float*)d_in[2];
  const float* mask   = (const float*)d_in[3];
  const float* ck     = (const float*)d_in[4];
  const float* cv     = (const float*)d_in[5];
  const float* wq     = (const float*)d_in[6];
  const float* wk     = (const float*)d_in[7];
  const float* wv     = (const float*)d_in[8];
  const float* wo     = (const float*)d_in[9];
  const int*   pos_p  = (const int*)d_in[10];
  float*       out    = (float*)d_out;

  float* Y    = (float*)d_ws;                       // 32 x 8192
  float* qws  = Y   + (size_t)BB * YCOLS;           // 32 x 4096
  float* kws  = qws + (size_t)BB * QKN;             // 32 x 2048
  float* vws  = kws + (size_t)BB * KVN;             // 32 x 2048
  float* attn = vws + (size_t)BB * KVN;             // 32 x 4096

  // QKV projections: tiles = (32/16) * (N/16), 8 tiles (waves) per block.
  gemm32_wmma_f32<QKN, HIDW, YCOLS>
      <<<(2 * (QKN / 16)) / 8, 256, 0, stream>>>(hidden, wq, Y);
  gemm32_wmma_f32<KVN, HIDW, YCOLS>
      <<<(2 * (KVN / 16)) / 8, 256, 0, stream>>>(hidden, wk, Y + QKN);
  gemm32_wmma_f32<KVN, HIDW, YCOLS>
      <<<(2 * (KVN / 16)) / 8, 256, 0, stream>>>(hidden, wv, Y + QKN + KVN);

  // RoPE + scatter
  rope_split<<<(BB * YCOLS) / 256, 256, 0, stream>>>(Y, fcos, fsin, qws, kws,
                                                     vws);

  // Attention: one block per (b, kv)
  attn_decode<<<BB * NKV, 256, 0, stream>>>(qws, kws, vws, ck, cv, mask, attn,
                                            pos_p);

  // Output projection: 32 x 4096 @ 4096 x 3072
  gemm32_wmma_f32<HIDW, QKN, HIDW>
      <<<(2 * (HIDW / 16)) / 8, 256, 0, stream>>>(attn, wo, out);
}